// SPGCC_26061861552728
// MI455X (gfx1250) — compile-verified
//
#include <hip/hip_runtime.h>

typedef __attribute__((ext_vector_type(2))) float v2f;
typedef __attribute__((ext_vector_type(8))) float v8f;

#define NODES 50000
#define NFEAT 512
#define NHID  128
#define NOUT  64

// ---------------------------------------------------------------------------
// C[M x N] = X[M x K] @ W[K x N], full fp32 via V_WMMA_F32_16X16X4_F32.
// One wave computes one 16x16 tile; block = 128 threads = 4 waves, each wave
// owning a different N-tile. No divergence anywhere -> EXEC all ones (WMMA
// requirement). M is a multiple of 16 (50000 = 3125*16), K a multiple of 4,
// N a multiple of 64, so no bounds checks are needed.
//
// ISA VGPR layouts (CDNA5 ISA):
//   A 16x4 f32 : lanes 0-15 -> M=lane, VGPR0=K0,VGPR1=K1; lanes 16-31 -> K2,K3
//   B 4x16 f32 : lanes 0-15 -> N=lane, VGPR0=K0,VGPR1=K1; lanes 16-31 -> K2,K3
//   C/D 16x16  : VGPR r, lanes 0-15 -> M=r, lanes 16-31 -> M=r+8
// ---------------------------------------------------------------------------
__global__ __launch_bounds__(128) void gemm_wmma_f32(
    const float* __restrict__ X, const float* __restrict__ W,
    float* __restrict__ C, int K, int N) {
  const int lane = threadIdx.x & 31;
  const int wave = threadIdx.x >> 5;
  const int l    = lane & 15;
  const int half = lane >> 4;               // 0: K0/K1 (M rows 0-7)  1: K2/K3
  const int m0   = blockIdx.x * 16;
  const int n0   = (blockIdx.y * 4 + wave) * 16;

  const float* xrow = X + (size_t)(m0 + l) * K + 2 * half;
  const float* wcol = W + (size_t)(2 * half) * N + (n0 + l);

  v8f acc = {};
#pragma unroll 4
  for (int k = 0; k < K; k += 4) {
    v2f a, b;
    a.x = xrow[k];
    a.y = xrow[k + 1];
    b.x = wcol[(size_t)k * N];
    b.y = wcol[(size_t)(k + 1) * N];
    // 8 args: (neg_a, A, neg_b, B, c_mod, C, reuse_a, reuse_b)
    acc = __builtin_amdgcn_wmma_f32_16x16x4_f32(false, a, false, b,
                                                (short)0, acc, false, false);
  }

  float* crow = C + (size_t)(m0 + 8 * half) * N + (n0 + l);
#pragma unroll
  for (int r = 0; r < 8; ++r)
    crow[(size_t)r * N] = acc[r];
}

// ---------------------------------------------------------------------------
// Seed the aggregation buffer with the broadcast bias so the SpMM atomics land
// directly on top of it (saves a finalize pass). W is a power of two.
// ---------------------------------------------------------------------------
__global__ __launch_bounds__(256) void init_bias(
    float* __restrict__ out, const float* __restrict__ b, int total, int W) {
  int i = blockIdx.x * blockDim.x + threadIdx.x;
  if (i < total) out[i] = b[i & (W - 1)];
}

// ---------------------------------------------------------------------------
// out[row[e], :] += val[e] * sup[col[e], :]
// One float4 feature chunk per thread; with W=128 a full wave covers one edge,
// so row/col/val loads are wave-uniform and the 512B gather is fully
// coalesced. The 25.6MB (or 12.8MB) support buffer fits in the 192MB L2, and
// the no-return f32 atomics resolve at L2 -> L2-bandwidth bound.
// ---------------------------------------------------------------------------
template <int W>
__global__ __launch_bounds__(256) void spmm_scatter(
    const int* __restrict__ row, const int* __restrict__ col,
    const float* __restrict__ val, const float* __restrict__ sup,
    float* __restrict__ out, int nEdges) {
  constexpr int W4 = W / 4;
  int gid = blockIdx.x * blockDim.x + threadIdx.x;
  int e   = gid / W4;                 // constexpr divisor -> shift
  int f   = (gid - e * W4) * 4;
  if (e >= nEdges) return;

  const int   r = row[e];
  const int   c = col[e];
  const float v = val[e];

  const float4 m = *(const float4*)(sup + (size_t)c * W + f);
  float* o = out + (size_t)r * W + f;
  __hip_atomic_fetch_add(o + 0, v * m.x, __ATOMIC_RELAXED, __HIP_MEMORY_SCOPE_AGENT);
  __hip_atomic_fetch_add(o + 1, v * m.y, __ATOMIC_RELAXED, __HIP_MEMORY_SCOPE_AGENT);
  __hip_atomic_fetch_add(o + 2, v * m.z, __ATOMIC_RELAXED, __HIP_MEMORY_SCOPE_AGENT);
  __hip_atomic_fetch_add(o + 3, v * m.w, __ATOMIC_RELAXED, __HIP_MEMORY_SCOPE_AGENT);
}

__global__ __launch_bounds__(256) void relu_ip(float* __restrict__ x, int n) {
  int i = blockIdx.x * blockDim.x + threadIdx.x;
  if (i < n) x[i] = fmaxf(x[i], 0.0f);
}

// ---------------------------------------------------------------------------
extern "C" void kernel_launch(void* const* d_in, const int* in_sizes, int n_in,
                              void* d_out, int out_size, void* d_ws, size_t ws_size,
                              hipStream_t stream) {
  const float* sp  = (const float*)d_in[0];
  const float* aug = (const float*)d_in[1];
  const int*   row = (const int*)d_in[2];
  const int*   col = (const int*)d_in[3];
  const float* val = (const float*)d_in[4];
  const float* w1  = (const float*)d_in[5];
  const float* b1  = (const float*)d_in[6];
  const float* w2  = (const float*)d_in[7];
  const float* b2  = (const float*)d_in[8];
  const float* w31 = (const float*)d_in[9];
  const float* b31 = (const float*)d_in[10];
  const float* w32 = (const float*)d_in[11];
  const float* b32 = (const float*)d_in[12];
  const int E = in_sizes[2];

  float* out  = (float*)d_out;
  float* bufA = (float*)d_ws;                       // NODES*NHID f32 (support)
  float* bufB = bufA + (size_t)NODES * NHID;        // NODES*NHID f32 (agg / h)

  const int nHidElems = NODES * NHID;               // 6.4M
  const int nOutElems = NODES * NOUT;               // 3.2M
  const dim3 gemmGridHid(NODES / 16, NHID / 64);    // (3125, 2)
  const dim3 gemmGridOut(NODES / 16, NOUT / 64);    // (3125, 1)
  const int eb128 = (int)(((long long)E * (NHID / 4) + 255) / 256);
  const int eb64  = (int)(((long long)E * (NOUT / 4) + 255) / 256);

  for (int br = 0; br < 2; ++br) {
    const float* x  = br ? aug : sp;
    float* o1 = out + (size_t)(2 * br + 0) * nOutElems;
    float* o2 = out + (size_t)(2 * br + 1) * nOutElems;

    // ---- layer 1: h1 = relu(spmm(x @ w1) + b1) ----
    gemm_wmma_f32<<<gemmGridHid, 128, 0, stream>>>(x, w1, bufA, NFEAT, NHID);
    init_bias<<<(nHidElems + 255) / 256, 256, 0, stream>>>(bufB, b1, nHidElems, NHID);
    spmm_scatter<NHID><<<eb128, 256, 0, stream>>>(row, col, val, bufA, bufB, E);
    relu_ip<<<(nHidElems + 255) / 256, 256, 0, stream>>>(bufB, nHidElems);

    // ---- layer 2: h2 = relu(spmm(h1 @ w2) + b2) ----
    gemm_wmma_f32<<<gemmGridHid, 128, 0, stream>>>(bufB, w2, bufA, NHID, NHID);
    init_bias<<<(nHidElems + 255) / 256, 256, 0, stream>>>(bufB, b2, nHidElems, NHID);
    spmm_scatter<NHID><<<eb128, 256, 0, stream>>>(row, col, val, bufA, bufB, E);
    relu_ip<<<(nHidElems + 255) / 256, 256, 0, stream>>>(bufB, nHidElems);

    // ---- head 1: out1 = spmm(h2 @ w31) + b31 ----
    gemm_wmma_f32<<<gemmGridOut, 128, 0, stream>>>(bufB, w31, bufA, NHID, NOUT);
    init_bias<<<(nOutElems + 255) / 256, 256, 0, stream>>>(o1, b31, nOutElems, NOUT);
    spmm_scatter<NOUT><<<eb64, 256, 0, stream>>>(row, col, val, bufA, o1, E);

    // ---- head 2: out2 = spmm(h2 @ w32) + b32 ----
    gemm_wmma_f32<<<gemmGridOut, 128, 0, stream>>>(bufB, w32, bufA, NHID, NOUT);
    init_bias<<<(nOutElems + 255) / 256, 256, 0, stream>>>(o2, b32, nOutElems, NOUT);
    spmm_scatter<NOUT><<<eb64, 256, 0, stream>>>(row, col, val, bufA, o2, E);
  }
}